// SRUCell_19731079758148
// MI455X (gfx1250) — compile-verified
//
#include <hip/hip_runtime.h>

typedef __attribute__((ext_vector_type(2))) float v2f;
typedef __attribute__((ext_vector_type(8))) float v8f;
typedef int v4i __attribute__((vector_size(4 * sizeof(int))));

#define GLOBAL_AS __attribute__((address_space(1)))
#define LDS_AS    __attribute__((address_space(3)))

typedef LDS_AS float*  lds_fptr;
typedef LDS_AS v4i*    lds_v4ptr;
typedef LDS_AS int*    lds_iptr;

#if __has_builtin(__builtin_amdgcn_global_load_async_to_lds_b128) && \
    __has_builtin(__builtin_amdgcn_global_load_async_to_lds_b32) &&  \
    __has_builtin(__builtin_amdgcn_s_wait_asynccnt)
#define USE_ASYNC_LDS 1
#else
#define USE_ASYNC_LDS 0
#endif

constexpr int L_SEQ = 1024;
constexpr int BATCH = 16;
constexpr int D_DIM = 1024;
constexpr int M_TOT = L_SEQ * BATCH;   // 16384
constexpr int N_TOT = 3 * D_DIM;       // 3072
constexpr int K_TOT = D_DIM;           // 1024

constexpr int BM = 128;
constexpr int BN = 128;
constexpr int BK = 16;

// ---------------------------------------------------------------------------
// GEMM: U[m,n] = sum_k X[m,k] * W[k,n]   (fp32, V_WMMA_F32_16X16X4_F32)
// Block tile 128x128, K-step 16, 8 waves (wave32), each wave owns 64x32.
// A staged [m][k]; B staged pair-interleaved [(k/2)][n][2] so every WMMA
// fragment is a contiguous b64 LDS read. Tiles move memory->LDS with CDNA5
// async copies (ASYNCcnt), double-buffered. All LDS destinations are AS(3)-
// typed up front so no addrspacecast null-guards appear in the K-loop.
// ---------------------------------------------------------------------------
__global__ __launch_bounds__(256)
void sru_gemm_wmma(const float* __restrict__ X, const float* __restrict__ W,
                   float* __restrict__ U) {
  __shared__ float As[2][BM * BK];            // [m][k], row stride 16 floats
  __shared__ float Bs[2][(BK / 2) * BN * 2];  // [(k/2)][n][2]

  const int tid  = threadIdx.x;
  const int wave = tid >> 5;
  const int lane = tid & 31;
  const int lr   = lane & 15;   // M/N index within 16
  const int hi   = lane >> 4;   // K-half select (A/B), M-half select (C/D)

  const int wm = wave & 1;      // 0..1 -> 64-row slice of block tile
  const int wn = wave >> 1;     // 0..3 -> 32-col slice of block tile

  const int m0 = blockIdx.y * BM;
  const int n0 = blockIdx.x * BN;

  // A staging: one b128 chunk per call site
  const int ar = tid >> 2;           // 0..63 (rows +0 and +64)
  const int ac = (tid & 3) * 4;      // 0,4,8,12
  // B staging: per-thread column bn; k = 2j + t7 for issues j=0..7
  const int t7 = tid >> 7;           // 0..1
  const int bn = tid & 127;          // 0..127

  v8f acc[4][2];
  #pragma unroll
  for (int i = 0; i < 4; ++i)
    #pragma unroll
    for (int j = 0; j < 2; ++j)
      acc[i][j] = v8f{};

  // running global pointers (advance by BK rows of W / BK cols of X per step)
  const float* aPtr = X + (size_t)(m0 + ar) * K_TOT + ac;
  const float* bPtr = W + (size_t)t7 * N_TOT + n0 + bn;

  // LDS destinations, AS(3)-typed once (addrspacecast hoisted out of the loop)
  lds_fptr aL0[2] = { (lds_fptr)&As[0][ar * BK + ac],
                      (lds_fptr)&As[1][ar * BK + ac] };
  lds_fptr aL1[2] = { (lds_fptr)&As[0][(ar + 64) * BK + ac],
                      (lds_fptr)&As[1][(ar + 64) * BK + ac] };
  lds_fptr bL[2]  = { (lds_fptr)&Bs[0][2 * bn + t7],
                      (lds_fptr)&Bs[1][2 * bn + t7] };

#if USE_ASYNC_LDS
  auto stage_tile = [&](const float* ap, const float* bp, int s) {
    __builtin_amdgcn_global_load_async_to_lds_b128(
        (GLOBAL_AS v4i*)ap,                        (lds_v4ptr)aL0[s], 0, 0);
    __builtin_amdgcn_global_load_async_to_lds_b128(
        (GLOBAL_AS v4i*)(ap + (size_t)64 * K_TOT), (lds_v4ptr)aL1[s], 0, 0);
    lds_fptr bd = bL[s];
    #pragma unroll
    for (int j = 0; j < 8; ++j)
      __builtin_amdgcn_global_load_async_to_lds_b32(
          (GLOBAL_AS int*)(bp + (size_t)j * 2 * N_TOT),
          (lds_iptr)(bd + j * (2 * BN)), 0, 0);
  };
  auto wait_stage = [&]() { __builtin_amdgcn_s_wait_asynccnt(0); };
#else
  auto stage_tile = [&](const float* ap, const float* bp, int s) {
    v4i a0 = *(const v4i*)(ap);
    v4i a1 = *(const v4i*)(ap + (size_t)64 * K_TOT);
    *(lds_v4ptr)aL0[s] = a0;
    *(lds_v4ptr)aL1[s] = a1;
    lds_fptr bd = bL[s];
    #pragma unroll
    for (int j = 0; j < 8; ++j)
      bd[j * (2 * BN)] = bp[(size_t)j * 2 * N_TOT];
  };
  auto wait_stage = [&]() {};
#endif

  stage_tile(aPtr, bPtr, 0);
  wait_stage();
  __syncthreads();

  const int NK = K_TOT / BK;   // 64 K-steps
  for (int kt = 0; kt < NK; ++kt) {
    const int s = kt & 1;
    aPtr += BK;
    bPtr += (size_t)BK * N_TOT;
    if (kt + 1 < NK) stage_tile(aPtr, bPtr, s ^ 1);  // overlaps WMMA below

    // Preload all B fragments for this K-tile (reused by all 4 mt rows).
    v2f bf[4][2];
    #pragma unroll
    for (int k4 = 0; k4 < 4; ++k4) {
      const int kh = k4 * 2 + hi;   // (kk>>1), kk = k4*4 + hi*2
      #pragma unroll
      for (int nt = 0; nt < 2; ++nt)
        bf[k4][nt] = *(const v2f*)&Bs[s][kh * (2 * BN) + 2 * (wn * 32 + nt * 16 + lr)];
    }

    #pragma unroll
    for (int k4 = 0; k4 < 4; ++k4) {
      const int kk = k4 * 4 + hi * 2;  // lanes 0-15: K={0,1}; lanes 16-31: {2,3}
      v2f af[4];
      #pragma unroll
      for (int mt = 0; mt < 4; ++mt)
        af[mt] = *(const v2f*)&As[s][(wm * 64 + mt * 16 + lr) * BK + kk];
      #pragma unroll
      for (int mt = 0; mt < 4; ++mt)
        #pragma unroll
        for (int nt = 0; nt < 2; ++nt)
          acc[mt][nt] = __builtin_amdgcn_wmma_f32_16x16x4_f32(
              false, af[mt], false, bf[k4][nt], (short)0, acc[mt][nt], false, false);
    }

    if (kt + 1 < NK) wait_stage();
    __syncthreads();
  }

  // C/D layout: VGPR v holds M = hi*8 + v, N = lane&15
  #pragma unroll
  for (int mt = 0; mt < 4; ++mt) {
    #pragma unroll
    for (int nt = 0; nt < 2; ++nt) {
      const int n = n0 + wn * 32 + nt * 16 + lr;
      #pragma unroll
      for (int v = 0; v < 8; ++v) {
        const int m = m0 + wm * 64 + mt * 16 + hi * 8 + v;
        U[(size_t)m * N_TOT + n] = acc[mt][nt][v];
      }
    }
  }
}

// ---------------------------------------------------------------------------
// Scan: one thread per (b,d); serial over L. Gate k of feature d is column
// 3*d+k of U. Prefetch next L-step (192KB stride) to hide latency in the
// serial chain. h and c_last written to d_out in tuple order.
// ---------------------------------------------------------------------------
__global__ __launch_bounds__(256)
void sru_scan(const float* __restrict__ U, const float* __restrict__ X,
              const float* __restrict__ bias, const float* __restrict__ c0,
              float* __restrict__ H, float* __restrict__ Clast) {
  const int g = blockIdx.x * 256 + threadIdx.x;  // 0..16383
  const int b = g >> 10;
  const int d = g & (D_DIM - 1);

  const float b1 = bias[d];
  const float b2 = bias[D_DIM + d];
  float c = c0[b * D_DIM + d];

  const float* up = U + (size_t)b * N_TOT + 3 * d;   // row l*B+b
  const float* xp = X + (size_t)b * D_DIM + d;
  float*       hp = H + (size_t)b * D_DIM + d;

  for (int l = 0; l < L_SEQ; ++l) {
    __builtin_prefetch(up + (size_t)BATCH * N_TOT, 0, 1);
    __builtin_prefetch(xp + (size_t)BATCH * D_DIM, 0, 1);
    const float u0 = up[0];
    const float z1 = up[1] + b1;
    const float z2 = up[2] + b2;
    const float g1 = 1.0f / (1.0f + __expf(-z1));
    const float g2 = 1.0f / (1.0f + __expf(-z2));
    c = (c - u0) * g1 + u0;
    const float xt = *xp;
    const float e2c = __expf(2.0f * c);
    const float th  = 1.0f - 2.0f / (e2c + 1.0f);   // tanh(c), inf-safe
    *hp = (th - xt) * g2 + xt;
    up += (size_t)BATCH * N_TOT;
    xp += (size_t)BATCH * D_DIM;
    hp += (size_t)BATCH * D_DIM;
  }
  Clast[b * D_DIM + d] = c;
}

// ---------------------------------------------------------------------------
extern "C" void kernel_launch(void* const* d_in, const int* in_sizes, int n_in,
                              void* d_out, int out_size, void* d_ws, size_t ws_size,
                              hipStream_t stream) {
  const float* x    = (const float*)d_in[0];   // (L, B, D)
  const float* w    = (const float*)d_in[1];   // (D, 3D)
  const float* bias = (const float*)d_in[2];   // (2D)
  const float* c0   = (const float*)d_in[3];   // (B, D)

  float* h     = (float*)d_out;                        // (L, B, D)
  float* clast = h + (size_t)L_SEQ * BATCH * D_DIM;    // (B, D)
  float* U     = (float*)d_ws;                         // (L*B, 3D) = 192 MB scratch

  dim3 gemmGrid(N_TOT / BN, M_TOT / BM);   // (24, 128)
  sru_gemm_wmma<<<gemmGrid, 256, 0, stream>>>(x, w, U);

  sru_scan<<<(BATCH * D_DIM) / 256, 256, 0, stream>>>(U, x, bias, c0, h, clast);
}